// SampleAndGroup_37744172597321
// MI455X (gfx1250) — compile-verified
//
#include <hip/hip_runtime.h>
#include <math.h>

// Problem constants (from reference)
#define B_       8
#define N_       8192
#define C_       64
#define NPOINT_  2048
#define NSAMPLE_ 32
#define RADIUS2_ 0.04f
#define OUTCH_   (3 + C_)   // 67

typedef __attribute__((ext_vector_type(2))) float v2f;
typedef __attribute__((ext_vector_type(8))) float v8f;

// ---------------------------------------------------------------------------
// Kernel A: farthest point sampling. One block per batch (serial chain over
// 2048 steps). Points live in registers (8 per thread), argmax via wave32
// shuffle reduce + 32-entry LDS combine. Writes new_xyz directly into d_out.
// ---------------------------------------------------------------------------
__global__ __launch_bounds__(1024) void fps_kernel(const float* __restrict__ xyz,
                                                   float* __restrict__ out) {
  const int b    = blockIdx.x;
  const int tid  = threadIdx.x;
  const int lane = tid & 31;
  const int wave = tid >> 5;
  const float* xb = xyz + (size_t)b * N_ * 3;
  float* new_xyz  = out + (size_t)b * NPOINT_ * 3;

  float px[8], py[8], pz[8], dist[8];
#pragma unroll
  for (int j = 0; j < 8; ++j) {
    int p = tid + j * 1024;
    px[j] = xb[p * 3 + 0];
    py[j] = xb[p * 3 + 1];
    pz[j] = xb[p * 3 + 2];
    dist[j] = INFINITY;
  }

  __shared__ float s_wd[32];
  __shared__ int   s_wi[32];
  __shared__ float s_c[3];
  __shared__ int   s_far;

  if (tid == 0) { s_c[0] = px[0]; s_c[1] = py[0]; s_c[2] = pz[0]; }  // far=0 start

  for (int s = 0; s < NPOINT_; ++s) {
    __syncthreads();
    const float cx = s_c[0], cy = s_c[1], cz = s_c[2];
    if (tid == 0) {
      new_xyz[s * 3 + 0] = cx;
      new_xyz[s * 3 + 1] = cy;
      new_xyz[s * 3 + 2] = cz;
    }

    float bd = -1.0f;
    int   bi = 0;
#pragma unroll
    for (int j = 0; j < 8; ++j) {
      float dx = px[j] - cx, dy = py[j] - cy, dz = pz[j] - cz;
      float d  = dx * dx + dy * dy + dz * dz;
      float nd = fminf(dist[j], d);
      dist[j] = nd;
      int p = tid + j * 1024;
      if (nd > bd) { bd = nd; bi = p; }   // ascending p => first-max kept
    }
    // wave32 argmax reduce (prefer lower index on ties, like jnp.argmax)
#pragma unroll
    for (int off = 16; off > 0; off >>= 1) {
      float od = __shfl_down(bd, off, 32);
      int   oi = __shfl_down(bi, off, 32);
      if (od > bd || (od == bd && oi < bi)) { bd = od; bi = oi; }
    }
    if (lane == 0) { s_wd[wave] = bd; s_wi[wave] = bi; }
    __syncthreads();
    if (wave == 0) {
      bd = s_wd[lane];
      bi = s_wi[lane];
#pragma unroll
      for (int off = 16; off > 0; off >>= 1) {
        float od = __shfl_down(bd, off, 32);
        int   oi = __shfl_down(bi, off, 32);
        if (od > bd || (od == bd && oi < bi)) { bd = od; bi = oi; }
      }
      if (lane == 0) s_far = bi;
    }
    __syncthreads();
    const int far = s_far;
    if (tid == (far & 1023)) {            // owner thread broadcasts coords
      int j = far >> 10;
      s_c[0] = px[j]; s_c[1] = py[j]; s_c[2] = pz[j];
    }
  }
}

// ---------------------------------------------------------------------------
// Kernel B: ball query via V_WMMA_F32_16X16X4_F32 + ballot-based selection.
//   dist^2(q_m, p_n) = |q|^2 - 2 q.p + |p|^2
//   A (16x4, rows=queries) : [qx, qy, qz, |q|^2]   (loop-invariant!)
//   B (4x16, cols=points)  : [-2px, -2py, -2pz, 1]
//   C (16x16)              : |p_n|^2 broadcast along M (per-lane constant)
// D lane L, vgpr r = dist^2(query r+8*(L/16), point L%16), so one
// __ballot(d[r] < r^2) yields the in-ball bitmask of 16 points for TWO query
// rows at once -- no LDS tile, no barriers in the hot loop. Owner lanes drain
// their mask with ctz in ascending index order (PointNet++ first-32).
// Mask routing uses a FLAT select chain with loop-invariant lane predicates
// so it lowers to hoisted v_cmp + per-tile v_cndmask (no EXEC branch blocks).
// ---------------------------------------------------------------------------
__device__ __forceinline__ void ballq_tile(const float* __restrict__ xb, int base,
                                           int h, int n, int L, int q, v2f a,
                                           int* idxbuf, int& cnt) {
  const int pi = base + n;
  const float ppx = xb[pi * 3 + 0];
  const float ppy = xb[pi * 3 + 1];
  const float ppz = xb[pi * 3 + 2];
  const float pn2 = ppx * ppx + ppy * ppy + ppz * ppz;
  v2f bm;                                // B operand: lane holds K=2h, 2h+1
  bm.x = h ? (-2.0f * ppz) : (-2.0f * ppx);
  bm.y = h ? 1.0f          : (-2.0f * ppy);
  v8f c;                                 // C: |p_n|^2, constant along M
#pragma unroll
  for (int r = 0; r < 8; ++r) c[r] = pn2;

  v8f d = __builtin_amdgcn_wmma_f32_16x16x4_f32(
      /*neg_a=*/false, a, /*neg_b=*/false, bm,
      /*c_mod=*/(short)0, c, /*reuse_a=*/false, /*reuse_b=*/false);

  // Route each wave-uniform ballot mask to its owner lanes via a flat
  // cndmask chain; the (q&7)==r compares are loop-invariant (hoisted).
  unsigned ma = 0u;
#pragma unroll
  for (int r = 0; r < 8; ++r) {
    const unsigned mkr = (unsigned)__ballot(d[r] < RADIUS2_);
    ma = ((q & 7) == r) ? mkr : ma;
  }
  unsigned mm = (q < 8) ? (ma & 0xFFFFu) : (ma >> 16);

  if (L < 16) {
    while (mm && cnt < NSAMPLE_) {       // usually empty (~3.3% density)
      int m = __builtin_ctz(mm);
      mm &= mm - 1;
      idxbuf[q * NSAMPLE_ + cnt] = base + m;
      ++cnt;
    }
  }
}

__global__ __launch_bounds__(32) void ballq_kernel(const float* __restrict__ xyz,
                                                   const float* __restrict__ out,
                                                   int* __restrict__ idx_ws) {
  const int blk = blockIdx.x;           // B * (NPOINT/16) blocks
  const int b   = blk >> 7;             // / 128
  const int q0  = (blk & 127) << 4;     // * 16
  const int L   = threadIdx.x;
  const int n   = L & 15;
  const int h   = L >> 4;
  const int q   = n;                    // query owned by lanes 0..15

  const float* xb = xyz + (size_t)b * N_ * 3;
  const float* nq = out + ((size_t)b * NPOINT_ + q0) * 3;   // new_xyz slice

  const float qx = nq[n * 3 + 0];
  const float qy = nq[n * 3 + 1];
  const float qz = nq[n * 3 + 2];
  const float qn2 = qx * qx + qy * qy + qz * qz;
  v2f a;                                 // A operand: loop-invariant
  a.x = h ? qz  : qx;
  a.y = h ? qn2 : qy;

  __shared__ int idxbuf[16 * NSAMPLE_];
  int cnt = 0;

  for (int t = 0; t < N_ / 16; t += 4) {
    __builtin_prefetch(xb + ((t + 4) * 16 + n) * 3, 0, 3);  // global_prefetch_b8
    ballq_tile(xb, (t + 0) * 16, h, n, L, q, a, idxbuf, cnt);
    ballq_tile(xb, (t + 1) * 16, h, n, L, q, a, idxbuf, cnt);
    ballq_tile(xb, (t + 2) * 16, h, n, L, q, a, idxbuf, cnt);
    ballq_tile(xb, (t + 3) * 16, h, n, L, q, a, idxbuf, cnt);
    const int done = (L < 16) ? (cnt >= NSAMPLE_) : 1;
    if (__all(done)) break;              // uniform exit: EXEC all-1s at WMMA
  }

  if (L < 16) {                          // pad with first in-ball index
    const int c0 = (cnt > 0) ? idxbuf[q * NSAMPLE_] : 0;
    for (int k = cnt; k < NSAMPLE_; ++k) idxbuf[q * NSAMPLE_ + k] = c0;
  }
  __syncthreads();

  int* dst = idx_ws + ((size_t)b * NPOINT_ + q0) * NSAMPLE_;
  for (int i = L; i < 16 * NSAMPLE_; i += 32) dst[i] = idxbuf[i];
}

// ---------------------------------------------------------------------------
// Kernel C: gather + group. 140 MB streaming output => pure bandwidth
// (~6-12 us floor at 23.3 TB/s). Flat element index keeps both the output
// writes and the `points` gathers coalesced along the channel dimension.
// ---------------------------------------------------------------------------
__global__ __launch_bounds__(256) void group_kernel(const float* __restrict__ xyz,
                                                    const float* __restrict__ points,
                                                    const int* __restrict__ idx_ws,
                                                    float* __restrict__ out) {
  const long long total = (long long)B_ * NPOINT_ * NSAMPLE_ * OUTCH_;
  const float* new_xyz    = out;
  float*       new_points = out + (size_t)B_ * NPOINT_ * 3;

  for (long long o = blockIdx.x * (long long)blockDim.x + threadIdx.x; o < total;
       o += (long long)gridDim.x * blockDim.x) {
    const int       c  = (int)(o % OUTCH_);
    const long long g  = o / OUTCH_;          // flat (b, s, k)
    const long long bs = g / NSAMPLE_;
    const int       s  = (int)(bs % NPOINT_);
    const int       b  = (int)(bs / NPOINT_);

    const int id = idx_ws[g];
    __builtin_prefetch(idx_ws + g + 256, 0, 3);   // global_prefetch_b8

    float v;
    if (c < 3) {
      v = xyz[((size_t)b * N_ + id) * 3 + c] -
          new_xyz[((size_t)b * NPOINT_ + s) * 3 + c];
    } else {
      v = points[((size_t)b * N_ + id) * C_ + (c - 3)];
    }
    new_points[o] = v;
  }
}

// ---------------------------------------------------------------------------
extern "C" void kernel_launch(void* const* d_in, const int* in_sizes, int n_in,
                              void* d_out, int out_size, void* d_ws, size_t ws_size,
                              hipStream_t stream) {
  const float* xyz    = (const float*)d_in[0];   // (B, N, 3)  f32
  const float* points = (const float*)d_in[1];   // (B, N, C)  f32
  float* out   = (float*)d_out;                  // new_xyz ++ new_points (f32)
  int*   idxws = (int*)d_ws;                     // (B, NPOINT, NSAMPLE) int32, 2 MB

  // 1) FPS -> new_xyz (first 49152 floats of d_out)
  fps_kernel<<<B_, 1024, 0, stream>>>(xyz, out);

  // 2) Ball query (WMMA distance tiles + ballot select) -> index table in ws
  ballq_kernel<<<B_ * (NPOINT_ / 16), 32, 0, stream>>>(xyz, out, idxws);

  // 3) Gather + concat -> new_points
  group_kernel<<<2048, 256, 0, stream>>>(xyz, points, idxws, out);
}